// IntegerNeuron_84842783965746
// MI455X (gfx1250) — compile-verified
//
#include <hip/hip_runtime.h>

// IntegerNeuron temporal scan, MI455X (gfx1250).
// Memory-bound streaming scan: 256MB in + 256MB out -> ~22us roofline @23.3TB/s.
// Data path: global_load_async_to_lds_b128 (ASYNCcnt-tracked, 8 deep) -> LDS ->
// register scan -> nontemporal B128 stores.

typedef float v4f __attribute__((ext_vector_type(4)));

#define T_STEPS 8
#define CH      256
#define HW      1024                       // H*W = 32*32
#define NSP     8388608ull                 // B*C*H*W = 32*256*32*32
#define EPSF    1e-12f

// Wait until per-wave ASYNCcnt <= n (literal immediate folded into the asm string).
#define WAIT_ASYNCCNT(n) asm volatile("s_wait_asynccnt " #n ::: "memory")

__global__ __launch_bounds__(256) void
IntegerNeuron_scan_kernel(const float* __restrict__ x,
                          const float* __restrict__ prev_scale,
                          const float* __restrict__ prev_bias,
                          const float* __restrict__ vth,
                          const float* __restrict__ tau_p,
                          const int*   __restrict__ ifl_p,
                          float* __restrict__ out)
{
    // LDS staging: [T][256 threads][4 floats] = 32 KB. Each wave touches only
    // its own tid range -> no cross-wave sharing, no barrier required.
    __shared__ float lds[T_STEPS * 256 * 4];

    const int      tid = threadIdx.x;
    const unsigned blk = blockIdx.x;
    // One block == one full H*W slice of one (b, c): channel is block-uniform.
    const unsigned i4 = blk * (unsigned)HW + (unsigned)tid * 4u;
    const int      c  = (int)(blk & (CH - 1));

    // Block-uniform parameter math (scalarizes to SMEM loads + SALU).
    const float tau    = tau_p[0];
    const int   ifl    = ifl_p[0];
    const float inv    = tau / (prev_scale[c] + EPSF);
    const float bias_s = __builtin_rintf(prev_bias[c] * inv);  // RNE == jnp.round
    const float vth_s  = __builtin_rintf(vth[c] * inv);
    const float dm     = ifl ? 1.0f : tau;                      // drive multiplier

    // ---- Issue all 8 timestep loads as async global->LDS B128 (gfx1250) ----
    // 8 outstanding ASYNCcnt ops per wave = max MLP with zero VGPR pressure.
    const unsigned long long gbase = (unsigned long long)(const void*)x;
#pragma unroll
    for (int t = 0; t < T_STEPS; ++t) {
        // Generic shared-pointer low 32 bits == LDS byte address (flat aperture).
        const unsigned lds_addr =
            (unsigned)(size_t)(const void*)&lds[((unsigned)t * 256u + (unsigned)tid) * 4u];
        const unsigned long long gaddr =
            gbase + ((unsigned long long)t * NSP + (unsigned long long)i4) * 4ull;
        asm volatile("global_load_async_to_lds_b128 %0, %1, off"
                     :: "v"(lds_addr), "v"(gaddr)
                     : "memory");
    }

    // ---- Register-carried scan; drain loads in completion order ----
    float m0 = 0.f, m1 = 0.f, m2 = 0.f, m3 = 0.f;

    auto consume = [&](int t) {
#pragma clang fp contract(off)   // match reference's separate mul/add rounding
        const v4f xv = *(const v4f*)(&lds[((unsigned)t * 256u + (unsigned)tid) * 4u]);
        float s0, s1, s2, s3;
        m0 = (m0 + xv.x * dm) + bias_s; s0 = (m0 >= vth_s) ? 1.f : 0.f; m0 -= s0 * vth_s;
        m1 = (m1 + xv.y * dm) + bias_s; s1 = (m1 >= vth_s) ? 1.f : 0.f; m1 -= s1 * vth_s;
        m2 = (m2 + xv.z * dm) + bias_s; s2 = (m2 >= vth_s) ? 1.f : 0.f; m2 -= s2 * vth_s;
        m3 = (m3 + xv.w * dm) + bias_s; s3 = (m3 >= vth_s) ? 1.f : 0.f; m3 -= s3 * vth_s;
        v4f ov; ov.x = s0; ov.y = s1; ov.z = s2; ov.w = s3;
        // Spikes are written once and never re-read: non-temporal B128 store.
        __builtin_nontemporal_store(
            ov, (v4f*)(out + (unsigned long long)t * NSP + (unsigned long long)i4));
    };

    // Loads return "done" in order => ASYNCcnt <= 7-t means slot t is in LDS.
    WAIT_ASYNCCNT(7); consume(0);
    WAIT_ASYNCCNT(6); consume(1);
    WAIT_ASYNCCNT(5); consume(2);
    WAIT_ASYNCCNT(4); consume(3);
    WAIT_ASYNCCNT(3); consume(4);
    WAIT_ASYNCCNT(2); consume(5);
    WAIT_ASYNCCNT(1); consume(6);
    WAIT_ASYNCCNT(0); consume(7);
}

extern "C" void kernel_launch(void* const* d_in, const int* in_sizes, int n_in,
                              void* d_out, int out_size, void* d_ws, size_t ws_size,
                              hipStream_t stream) {
    const float* x          = (const float*)d_in[0];  // [T,B,C,H,W] f32
    const float* prev_scale = (const float*)d_in[1];  // [C]
    const float* prev_bias  = (const float*)d_in[2];  // [C]
    const float* vth        = (const float*)d_in[3];  // [C]
    const float* tau        = (const float*)d_in[4];  // [1]
    const int*   ifl        = (const int*)  d_in[5];  // scalar
    float*       out        = (float*)d_out;          // [T,B,C,H,W] f32

    (void)in_sizes; (void)n_in; (void)out_size; (void)d_ws; (void)ws_size;

    const unsigned blocks = (unsigned)(NSP / HW);     // 8192 blocks, 256 thr = 8 waves
    IntegerNeuron_scan_kernel<<<dim3(blocks), dim3(256), 0, stream>>>(
        x, prev_scale, prev_bias, vth, tau, ifl, out);
}